// FastSecureAttentionSystem_54992761258671
// MI455X (gfx1250) — compile-verified
//
#include <hip/hip_runtime.h>

typedef __attribute__((ext_vector_type(2))) float v2f;
typedef __attribute__((ext_vector_type(8))) float v8f;

#define D 7
#define S 512
#define TILE 16
#define NTILES 32          // S / TILE
#define WAVES 4
#define THREADS 128
#define TPW 8              // column tiles per wave
#define SS 520             // scores LDS row stride (floats), avoids bank conflicts

// LDS layout (floats):
// proj  [512*7]   @ 0      (3584)
// bias  [512*7]   @ 3584   (3584)
// sc    [16*520]  @ 7168   (8320)
// prm   [80]      @ 15488  (W:0..48, b:49..55, ab:56..62, gamma:63..69, beta:70..76)
// red   [128]     @ 15568
// rstat [32]      @ 15696  (rowmax[16], rowinv[16])
#define LDS_FLOATS 15728

__global__ __launch_bounds__(THREADS)
void fast_attn_kernel(const float* __restrict__ f, const float* __restrict__ W,
                      const float* __restrict__ bvec, const float* __restrict__ ab,
                      const float* __restrict__ gam, const float* __restrict__ bet,
                      float* __restrict__ out_att, float* __restrict__ out_w)
{
    __shared__ float lds[LDS_FLOATS];
    float* proj = lds;
    float* bia  = lds + 3584;
    float* sc   = lds + 7168;
    float* prm  = lds + 15488;
    float* red  = lds + 15568;
    float* rst  = lds + 15696;

    const int tid   = threadIdx.x;
    const int batch = blockIdx.x >> 5;
    const int tile  = blockIdx.x & 31;
    const int rowBase = tile * TILE;            // row tile start within batch
    const long tokBase = (long)batch * S;

    // ---- load params to LDS ----
    if (tid < 49)      prm[tid] = W[tid];
    else if (tid < 56) prm[tid] = bvec[tid - 49];
    else if (tid < 63) prm[tid] = ab[tid - 56];
    else if (tid < 70) prm[tid] = gam[tid - 63];
    else if (tid < 77) prm[tid] = bet[tid - 70];
    __syncthreads();

    // ---- biased + projected for all 512 tokens of this batch ----
    for (int t = tid; t < S; t += THREADS) {
        float x[D];
        const float* fp = f + (tokBase + t) * D;
        #pragma unroll
        for (int d = 0; d < D; ++d) {
            x[d] = fp[d] + prm[56 + d];
            bia[t * D + d] = x[d];
        }
        #pragma unroll
        for (int e = 0; e < D; ++e) {
            float acc = prm[49 + e];
            #pragma unroll
            for (int d = 0; d < D; ++d) acc += x[d] * prm[e * D + d];
            proj[t * D + e] = acc;
        }
    }
    __syncthreads();

    // ---- scores via V_WMMA_F32_16X16X4_F32 (K = 7, padded to 8) ----
    const int wave = tid >> 5;
    const int lane = tid & 31;
    const int m    = lane & 15;
    const int ks   = lane >> 4;             // 0: K pair {0,1}; 1: K pair {2,3}
    const float kmask = ks ? 0.0f : 1.0f;   // zeroes the K=7 pad lane (no branch)

    const int pr = (rowBase + m) * D;
    v2f a0, a1;
    a0.x = proj[pr + 2 * ks];      a0.y = proj[pr + 2 * ks + 1];
    a1.x = proj[pr + 4 + 2 * ks];  a1.y = proj[pr + 5] * kmask;   // K=5 or K=7->0

    const float scale = 0.3779644730092272f;   // 1/sqrt(7)

    for (int jj = 0; jj < TPW; ++jj) {
        const int j  = wave * TPW + jj;
        const int pc = (j * 16 + m) * D;
        v2f b0, b1;
        b0.x = proj[pc + 2 * ks];      b0.y = proj[pc + 2 * ks + 1];
        b1.x = proj[pc + 4 + 2 * ks];  b1.y = proj[pc + 5] * kmask;
        v8f c = {0.f, 0.f, 0.f, 0.f, 0.f, 0.f, 0.f, 0.f};
        c = __builtin_amdgcn_wmma_f32_16x16x4_f32(false, a0, false, b0, (short)0, c, false, false);
        c = __builtin_amdgcn_wmma_f32_16x16x4_f32(false, a1, false, b1, (short)0, c, false, false);
        #pragma unroll
        for (int r = 0; r < 8; ++r)
            sc[(r + 8 * ks) * SS + j * 16 + m] = c[r] * scale;
    }
    __syncthreads();

    // ---- softmax: rowwise max ----
    {
        const int row = tid >> 3, sub = tid & 7;
        float mx = -1e30f;
        for (int k = 0; k < 64; ++k) mx = fmaxf(mx, sc[row * SS + sub + 8 * k]);
        red[tid] = mx;
    }
    __syncthreads();
    if (tid < 16) {
        float mx = red[tid * 8];
        #pragma unroll
        for (int i = 1; i < 8; ++i) mx = fmaxf(mx, red[tid * 8 + i]);
        rst[tid] = mx;
    }
    __syncthreads();
    // ---- exp + rowwise sum (exp kept in LDS) ----
    {
        const int row = tid >> 3, sub = tid & 7;
        const float mx = rst[row];
        float s = 0.f;
        for (int k = 0; k < 64; ++k) {
            const int idx = row * SS + sub + 8 * k;
            const float e = __expf(sc[idx] - mx);
            sc[idx] = e;
            s += e;
        }
        red[tid] = s;
    }
    __syncthreads();
    if (tid < 16) {
        float s = 0.f;
        #pragma unroll
        for (int i = 0; i < 8; ++i) s += red[tid * 8 + i];
        rst[16 + tid] = 1.0f / s;
    }
    __syncthreads();

    // ---- stream weights to HBM (dominant traffic; non-temporal, coalesced) ----
    {
        float* wout = out_w + (tokBase + rowBase) * (long)S;
        for (int r = 0; r < TILE; ++r) {
            const float inv = rst[16 + r];
            for (int t = tid; t < S; t += THREADS)
                __builtin_nontemporal_store(sc[r * SS + t] * inv, &wout[(long)r * S + t]);
        }
    }

    // ---- attended = softmax(scores) @ biased, then layernorm ----
    if (tid < TILE * D) {                      // 112 lanes: (row, dim)
        const int r = tid / D, d = tid % D;
        float acc = 0.f;
        for (int t = 0; t < S; ++t)
            acc += sc[r * SS + t] * bia[t * D + d];
        red[tid] = acc * rst[16 + r];
    }
    __syncthreads();
    if (tid < TILE) {
        float v[D];
        float mu = 0.f;
        #pragma unroll
        for (int i = 0; i < D; ++i) { v[i] = red[tid * D + i]; mu += v[i]; }
        mu *= (1.0f / D);
        float var = 0.f;
        #pragma unroll
        for (int i = 0; i < D; ++i) { const float dd = v[i] - mu; var += dd * dd; }
        var *= (1.0f / D);
        const float rinv = rsqrtf(var + 1e-5f);
        float* ao = out_att + (tokBase + rowBase + tid) * D;
        #pragma unroll
        for (int i = 0; i < D; ++i)
            ao[i] = (v[i] - mu) * rinv * prm[63 + i] + prm[70 + i];
    }
}

extern "C" void kernel_launch(void* const* d_in, const int* in_sizes, int n_in,
                              void* d_out, int out_size, void* d_ws, size_t ws_size,
                              hipStream_t stream) {
    const float* f   = (const float*)d_in[0];
    const float* W   = (const float*)d_in[1];
    const float* b   = (const float*)d_in[2];
    const float* ab  = (const float*)d_in[3];
    const float* gam = (const float*)d_in[4];
    const float* bet = (const float*)d_in[5];

    float* out      = (float*)d_out;
    float* out_att  = out;                       // [256,512,7]
    float* out_w    = out + (long)256 * 512 * 7; // [256,512,512]

    dim3 grid(256 * NTILES);   // 8192 blocks: (batch, 16-row tile)
    dim3 block(THREADS);
    hipLaunchKernelGGL(fast_attn_kernel, grid, block, 0, stream,
                       f, W, b, ab, gam, bet, out_att, out_w);
}